// PatchMask_4492535792036
// MI455X (gfx1250) — compile-verified
//
#include <hip/hip_runtime.h>
#include <hip/hip_bf16.h>

// PatchMask: out[b,c,h,w] = x[b,c,h,w] * keep_mask[b, h/16, w/16]
//
// Memory-bound: 192 MiB in + 192 MiB out at 23.3 TB/s -> ~17 us floor.
// Strategy: each thread owns one 16-pixel row-span of one patch (64 B),
// so one mask byte feeds 4x global_load_b128 + 4x global_store_b128.
// Non-temporal hints since the 384 MiB stream exceeds the 192 MB L2.

typedef float f4 __attribute__((ext_vector_type(4)));

#define BB 64
#define CC 3
#define HH 512
#define WW 512
#define GROUPS (BB * CC * HH * (WW / 16))   // 3,145,728 threads, 16 px each
#define THREADS 256

__global__ __launch_bounds__(THREADS) void PatchMask_4492535792036_kernel(
    const f4* __restrict__ x,
    const unsigned char* __restrict__ keep,   // bool mask, 1 byte/elem [64,32,32]
    f4* __restrict__ out)
{
    unsigned idx16 = blockIdx.x * THREADS + threadIdx.x;   // 16-pixel group id
    if (idx16 >= GROUPS) return;

    // Decompose: per (b,c) there are 512*32 = 2^14 groups; 32 groups per row.
    unsigned pc = idx16 & 31u;                 // patch column (also 16-px group in row)
    unsigned h  = (idx16 >> 5) & 511u;         // pixel row
    unsigned bc = idx16 >> 14;                 // b*3 + c
    unsigned b  = bc / 3u;                     // -> s_mul_hi / v_mul_hi
    unsigned mi = b * 1024u + (h >> 4) * 32u + pc;

    float m = (float)keep[mi];                 // 0.0f or 1.0f (mask is cached; tiny)

    unsigned base = idx16 * 4u;                // float4 index; 64 B per thread

    // Issue all 4 B128 loads first (4 outstanding per lane), NT: stream, don't cache.
    f4 v0 = __builtin_nontemporal_load(&x[base + 0]);
    f4 v1 = __builtin_nontemporal_load(&x[base + 1]);
    f4 v2 = __builtin_nontemporal_load(&x[base + 2]);
    f4 v3 = __builtin_nontemporal_load(&x[base + 3]);

    v0 *= m;
    v1 *= m;
    v2 *= m;
    v3 *= m;

    __builtin_nontemporal_store(v0, &out[base + 0]);
    __builtin_nontemporal_store(v1, &out[base + 1]);
    __builtin_nontemporal_store(v2, &out[base + 2]);
    __builtin_nontemporal_store(v3, &out[base + 3]);
}

extern "C" void kernel_launch(void* const* d_in, const int* in_sizes, int n_in,
                              void* d_out, int out_size, void* d_ws, size_t ws_size,
                              hipStream_t stream) {
    const f4* x                 = (const f4*)d_in[0];
    const unsigned char* keep   = (const unsigned char*)d_in[1];
    f4* out                     = (f4*)d_out;

    const int grid = GROUPS / THREADS;   // 12288 blocks, exact
    PatchMask_4492535792036_kernel<<<grid, THREADS, 0, stream>>>(x, keep, out);
}